// SA_SKN_Res_encoder_4097398800730
// MI455X (gfx1250) — compile-verified
//
#include <hip/hip_runtime.h>
#include <cstdint>
#include <cstddef>

// ---------------------------------------------------------------------------
// Types / constants
// ---------------------------------------------------------------------------
typedef __attribute__((ext_vector_type(2))) float v2f;
typedef __attribute__((ext_vector_type(8))) float v8f;

#define ACT_NONE 0
#define ACT_RELU 1

static constexpr int BB = 8;          // batch
static constexpr int SHARE = 8;

__device__ __forceinline__ float actf(float x, int act) {
  return (act == ACT_RELU) ? fmaxf(x, 0.0f) : x;
}

static inline int cdiv(int a, int b) { return (a + b - 1) / b; }

// ---------------------------------------------------------------------------
// WMMA GEMM:  Y[b,o,n] = act( sum_c W[o,c]*X[b,c,n] + bias[o] + res[b,o,n] )
// 4 waves/block; each wave32 computes a 16x16 (OxN) tile via
// V_WMMA_F32_16X16X4_F32, K unrolled by 16 (4 WMMAs / iter).
// NN>0: compile-time N -> immediate-offset B loads off one running pointer.
// OOB rows/cols use clamped loads; only the K tail needs masking.
// ---------------------------------------------------------------------------
template <int NN>
__global__ void gemm_wmma_kernel(const float* __restrict__ W,
                                 const float* __restrict__ X,
                                 const float* __restrict__ bias,
                                 const float* __restrict__ res,
                                 float* __restrict__ Y,
                                 int O, int C, int Nrt, int act) {
  const int N = (NN > 0) ? NN : Nrt;
  const int wave = threadIdx.x >> 5;
  const int lane = threadIdx.x & 31;
  const int b    = blockIdx.z;
  const int o0   = blockIdx.y * 16;
  const int n0   = (blockIdx.x * (blockDim.x >> 5) + wave) * 16;
  if (n0 >= N) return;
  const int half = lane >> 4;
  const int l    = lane & 15;

  const int row   = o0 + l;
  const bool rowok = (row < O);
  const int rowc  = rowok ? row : (O - 1);
  const int col   = n0 + l;
  const bool colok = (col < N);
  const int colc  = colok ? col : (N - 1);

  const int kb = half * 2;   // this lane's K sub-offset within a 4-wide step
  const float* __restrict__ wp = W + (size_t)rowc * C + kb;
  const float* __restrict__ xq = X + (size_t)b * C * N + colc + (size_t)kb * N;

  v8f acc = {};
  const int C4 = C & ~3;
  int k0 = 0;

  // main loop: 16 K-values -> 4 WMMAs per iteration (unconditional loads)
  for (; k0 + 16 <= C4; k0 += 16) {
    __builtin_prefetch(xq + (size_t)16 * N, 0, 1);
    __builtin_prefetch(xq + (size_t)24 * N, 0, 1);
    v2f a0 = *(const v2f*)(wp + 0);
    v2f a1 = *(const v2f*)(wp + 4);
    v2f a2 = *(const v2f*)(wp + 8);
    v2f a3 = *(const v2f*)(wp + 12);
    v2f b0, b1, b2, b3;
    b0[0] = xq[(size_t)0 * N];
    b0[1] = xq[(size_t)1 * N];
    b1[0] = xq[(size_t)4 * N];
    b1[1] = xq[(size_t)5 * N];
    b2[0] = xq[(size_t)8 * N];
    b2[1] = xq[(size_t)9 * N];
    b3[0] = xq[(size_t)12 * N];
    b3[1] = xq[(size_t)13 * N];
    wp += 16;
    xq += (size_t)16 * N;
    acc = __builtin_amdgcn_wmma_f32_16x16x4_f32(false, a0, false, b0, (short)0, acc, false, false);
    acc = __builtin_amdgcn_wmma_f32_16x16x4_f32(false, a1, false, b1, (short)0, acc, false, false);
    acc = __builtin_amdgcn_wmma_f32_16x16x4_f32(false, a2, false, b2, (short)0, acc, false, false);
    acc = __builtin_amdgcn_wmma_f32_16x16x4_f32(false, a3, false, b3, (short)0, acc, false, false);
  }
  // K=4 steps
  for (; k0 < C4; k0 += 4) {
    v2f a = *(const v2f*)(wp);
    v2f bf;
    bf[0] = xq[(size_t)0 * N];
    bf[1] = xq[(size_t)1 * N];
    wp += 4;
    xq += (size_t)4 * N;
    acc = __builtin_amdgcn_wmma_f32_16x16x4_f32(false, a, false, bf, (short)0, acc, false, false);
  }
  // masked K tail (C not multiple of 4: C==3 or C==2 cases)
  if (k0 < C) {
    const float* __restrict__ Wrow = W + (size_t)rowc * C;
    const float* __restrict__ Xcol = X + (size_t)b * C * N + colc;
    v2f a, bf;
#pragma unroll
    for (int kk = 0; kk < 2; ++kk) {
      const int k = k0 + kb + kk;
      const bool ok = (k < C);
      const int kc = ok ? k : (C - 1);
      float av = Wrow[kc];
      float bv = Xcol[(size_t)kc * N];
      a[kk] = ok ? av : 0.0f;
      bf[kk] = ok ? bv : 0.0f;
    }
    acc = __builtin_amdgcn_wmma_f32_16x16x4_f32(false, a, false, bf, (short)0, acc, false, false);
  }

  float* __restrict__ Yb = Y + (size_t)b * O * N;
#pragma unroll
  for (int r = 0; r < 8; ++r) {
    const int o = o0 + r + half * 8;
    if (o < O && colok) {
      float v = acc[r];
      if (bias) v += bias[o];
      if (res)  v += res[((size_t)b * O + o) * N + col];
      Yb[(size_t)o * N + col] = actf(v, act);
    }
  }
}

// ---------------------------------------------------------------------------
// WMMA GEMM with fused neighbor gather on B:
//   Y[b,o, j*N+n] = act( sum_c W[o,c] * XA[b,c, idx[b,n*K+j]] + bias[o] )
// Columns = K*N; gathered source column resolved once per lane.
// ---------------------------------------------------------------------------
template <int NN>
__global__ void gemm_gather_kernel(const float* __restrict__ W,
                                   const float* __restrict__ XA,
                                   const int* __restrict__ idx,
                                   const float* __restrict__ bias,
                                   float* __restrict__ Y,
                                   int O, int C, int Nrt, int K, int act) {
  const int N = (NN > 0) ? NN : Nrt;
  const int wave = threadIdx.x >> 5;
  const int lane = threadIdx.x & 31;
  const int b    = blockIdx.z;
  const int o0   = blockIdx.y * 16;
  const int NC   = K * N;
  const int n0   = (blockIdx.x * (blockDim.x >> 5) + wave) * 16;
  if (n0 >= NC) return;
  const int half = lane >> 4;
  const int l    = lane & 15;

  const int row    = o0 + l;
  const bool rowok = (row < O);
  const int rowc   = rowok ? row : (O - 1);
  const int col    = n0 + l;
  const bool colok = (col < NC);
  const int colc   = colok ? col : (NC - 1);

  // resolve gathered source point for this lane's column
  const int j = colc / N;
  const int n = colc - j * N;
  const int src = idx[((size_t)b * N + n) * K + j];

  const int kb = half * 2;
  const float* __restrict__ wp = W + (size_t)rowc * C + kb;
  const float* __restrict__ xq = XA + (size_t)b * C * N + src + (size_t)kb * N;

  v8f acc = {};
  const int C4 = C & ~3;
  int k0 = 0;
  for (; k0 + 16 <= C4; k0 += 16) {
    __builtin_prefetch(xq + (size_t)16 * N, 0, 1);
    __builtin_prefetch(xq + (size_t)24 * N, 0, 1);
    v2f a0 = *(const v2f*)(wp + 0);
    v2f a1 = *(const v2f*)(wp + 4);
    v2f a2 = *(const v2f*)(wp + 8);
    v2f a3 = *(const v2f*)(wp + 12);
    v2f b0, b1, b2, b3;
    b0[0] = xq[(size_t)0 * N];
    b0[1] = xq[(size_t)1 * N];
    b1[0] = xq[(size_t)4 * N];
    b1[1] = xq[(size_t)5 * N];
    b2[0] = xq[(size_t)8 * N];
    b2[1] = xq[(size_t)9 * N];
    b3[0] = xq[(size_t)12 * N];
    b3[1] = xq[(size_t)13 * N];
    wp += 16;
    xq += (size_t)16 * N;
    acc = __builtin_amdgcn_wmma_f32_16x16x4_f32(false, a0, false, b0, (short)0, acc, false, false);
    acc = __builtin_amdgcn_wmma_f32_16x16x4_f32(false, a1, false, b1, (short)0, acc, false, false);
    acc = __builtin_amdgcn_wmma_f32_16x16x4_f32(false, a2, false, b2, (short)0, acc, false, false);
    acc = __builtin_amdgcn_wmma_f32_16x16x4_f32(false, a3, false, b3, (short)0, acc, false, false);
  }
  for (; k0 < C4; k0 += 4) {
    v2f a = *(const v2f*)(wp);
    v2f bf;
    bf[0] = xq[(size_t)0 * N];
    bf[1] = xq[(size_t)1 * N];
    wp += 4;
    xq += (size_t)4 * N;
    acc = __builtin_amdgcn_wmma_f32_16x16x4_f32(false, a, false, bf, (short)0, acc, false, false);
  }
  if (k0 < C) {
    const float* __restrict__ Wrow = W + (size_t)rowc * C;
    const float* __restrict__ Xcol = XA + (size_t)b * C * N + src;
    v2f a, bf;
#pragma unroll
    for (int kk = 0; kk < 2; ++kk) {
      const int k = k0 + kb + kk;
      const bool ok = (k < C);
      const int kc = ok ? k : (C - 1);
      float av = Wrow[kc];
      float bv = Xcol[(size_t)kc * N];
      a[kk] = ok ? av : 0.0f;
      bf[kk] = ok ? bv : 0.0f;
    }
    acc = __builtin_amdgcn_wmma_f32_16x16x4_f32(false, a, false, bf, (short)0, acc, false, false);
  }

  float* __restrict__ Yb = Y + (size_t)b * O * NC;
#pragma unroll
  for (int r = 0; r < 8; ++r) {
    const int o = o0 + r + half * 8;
    if (o < O && colok) {
      float v = acc[r];
      if (bias) v += bias[o];
      Yb[(size_t)o * NC + col] = actf(v, act);
    }
  }
}

// ---------------------------------------------------------------------------
// Elementwise / small kernels
// ---------------------------------------------------------------------------
__global__ void relu_kernel(float* dst, const float* src, int n) {
  int i = blockIdx.x * blockDim.x + threadIdx.x;
  if (i < n) dst[i] = fmaxf(src[i], 0.0f);
}

// dst(B,Ctot,N) rows [row0,row0+Csrc) <- src(B,Csrc,N)
__global__ void copy_rows_kernel(float* dst, const float* src,
                                 int Ctot, int row0, int Csrc, int N, int total) {
  int i = blockIdx.x * blockDim.x + threadIdx.x;
  if (i >= total) return;
  int n = i % N;
  int t = i / N;
  int c = t % Csrc;
  int b = t / Csrc;
  dst[((size_t)b * Ctot + row0 + c) * N + n] = src[((size_t)b * Csrc + c) * N + n];
}

// dst(B,Ctot,N) rows [row0,row0+Cv) <- broadcast vec(B,Cv) over N
__global__ void broadcast_rows_kernel(float* dst, const float* vec,
                                      int Ctot, int row0, int Cv, int N, int total) {
  int i = blockIdx.x * blockDim.x + threadIdx.x;
  if (i >= total) return;
  int n = i % N;
  int t = i / N;
  int c = t % Cv;
  int b = t / Cv;
  dst[((size_t)b * Ctot + row0 + c) * N + n] = vec[(size_t)b * Cv + c];
}

// wcat(B, rel*(K+1), N) = relu(concat(x1s (B,rel,N), x2s (B,rel,K,N) reshaped))
__global__ void pack_wcat_kernel(float* wcat, const float* x1s, const float* x2s,
                                 int rel, int K, int N, int total) {
  int i = blockIdx.x * blockDim.x + threadIdx.x;
  if (i >= total) return;
  const int P = rel * (K + 1);
  int n = i % N;
  int t = i / N;
  int p = t % P;
  int b = t / P;
  float v;
  if (p < rel) {
    v = x1s[((size_t)b * rel + p) * N + n];
  } else {
    int c2 = p - rel;
    int r = c2 / K;
    int j = c2 - r * K;
    v = x2s[((size_t)b * rel + r) * (size_t)(K * N) + (size_t)j * N + n];
  }
  wcat[((size_t)b * P + p) * N + n] = fmaxf(v, 0.0f);
}

// out(B,mid,N) = relu( sum_j wb[b, (m%ms)*K+j, n] * x3[b, m, j, n] )
__global__ void weighted_sum_kernel(float* out, const float* wb, const float* x3,
                                    int mid, int ms, int K, int N, int total) {
  int i = blockIdx.x * blockDim.x + threadIdx.x;
  if (i >= total) return;
  int n = i % N;
  int t = i / N;
  int m = t % mid;
  int b = t / mid;
  int mm = m % ms;
  const float* wp = wb + ((size_t)b * (K * ms) + (size_t)mm * K) * N + n;
  const float* xp = x3 + (((size_t)b * mid + m) * K) * N + n;
  float s = 0.0f;
  for (int j = 0; j < K; ++j) s += wp[(size_t)j * N] * xp[(size_t)j * N];
  out[((size_t)b * mid + m) * N + n] = fmaxf(s, 0.0f);
}

// softmax over 2 branches + weighted combine
__global__ void sk_combine_kernel(float* out, const float* f0, const float* f1,
                                  const float* vecs, int C, int N, int total) {
  int i = blockIdx.x * blockDim.x + threadIdx.x;
  if (i >= total) return;
  int t = i / N;
  int c = t % C;
  int b = t / C;
  float a0 = vecs[(size_t)b * 2 * C + c];
  float a1 = vecs[(size_t)b * 2 * C + C + c];
  float m = fmaxf(a0, a1);
  float e0 = __expf(a0 - m), e1 = __expf(a1 - m);
  float inv = 1.0f / (e0 + e1);
  out[i] = f0[i] * (e0 * inv) + f1[i] * (e1 * inv);
}

// fea_s[b,c] = mean_n (f0+f1)[b,c,n]
__global__ void mean_over_n_kernel(float* fea_s, const float* f0, const float* f1,
                                   int N, int total) {
  int i = blockIdx.x * blockDim.x + threadIdx.x;
  if (i >= total) return;
  const float* p0 = f0 + (size_t)i * N;
  const float* p1 = f1 + (size_t)i * N;
  float s = 0.0f;
  for (int n = 0; n < N; ++n) s += p0[n] + p1[n];
  fea_s[i] = s / (float)N;
}

// Y[b*ostride + obase + o] = act( X[b,:] . W[o,:] + bias[o] )
__global__ void linear_kernel(float* Y, const float* X, const float* W,
                              const float* bias, int IN, int OUT,
                              int ostride, int obase, int act, int total) {
  int i = blockIdx.x * blockDim.x + threadIdx.x;
  if (i >= total) return;
  int o = i % OUT;
  int b = i / OUT;
  float s = bias ? bias[o] : 0.0f;
  const float* x = X + (size_t)b * IN;
  const float* w = W + (size_t)o * IN;
  for (int c = 0; c < IN; ++c) s += x[c] * w[c];
  Y[(size_t)b * ostride + obase + o] = actf(s, act);
}

__global__ void max_over_n_kernel(float* out, const float* X, int N, int total) {
  int i = blockIdx.x * blockDim.x + threadIdx.x;
  if (i >= total) return;
  const float* p = X + (size_t)i * N;
  float m = p[0];
  for (int n = 1; n < N; ++n) m = fmaxf(m, p[n]);
  out[i] = m;
}

// ---------------------------------------------------------------------------
// Point kernels
// ---------------------------------------------------------------------------
// pts(B,N,3) <- features(B,3,N) first 3 rows
__global__ void transpose_pts_kernel(float* pts, const float* feat, int N, int total) {
  int i = blockIdx.x * blockDim.x + threadIdx.x;
  if (i >= total) return;
  int n = i % N;
  int b = i / N;
  for (int c = 0; c < 3; ++c)
    pts[((size_t)b * N + n) * 3 + c] = feat[((size_t)b * 3 + c) * N + n];
}

// K nearest (squared L2) neighbors of query(B,S,3) among all(B,N,3); K <= 20
__global__ void knn_query_kernel(int* idx, const float* all, const float* query,
                                 int N, int S, int K, int total) {
  int i = blockIdx.x * blockDim.x + threadIdx.x;
  if (i >= total) return;
  int s = i % S;
  int b = i / S;
  const float* q = query + ((size_t)b * S + s) * 3;
  const float qx = q[0], qy = q[1], qz = q[2];
  const float* P = all + (size_t)b * N * 3;
  float dbest[20];
  int ibest[20];
  for (int j = 0; j < K; ++j) { dbest[j] = 3.4e38f; ibest[j] = 0; }
  for (int m = 0; m < N; ++m) {
    float dx = P[m * 3 + 0] - qx;
    float dy = P[m * 3 + 1] - qy;
    float dz = P[m * 3 + 2] - qz;
    float d = dx * dx + dy * dy + dz * dz;
    if (d < dbest[K - 1]) {
      int pos = K - 1;
      while (pos > 0 && dbest[pos - 1] > d) {
        dbest[pos] = dbest[pos - 1];
        ibest[pos] = ibest[pos - 1];
        --pos;
      }
      dbest[pos] = d;
      ibest[pos] = m;
    }
  }
  int* op = idx + ((size_t)b * S + s) * K;
  for (int j = 0; j < K; ++j) op[j] = ibest[j];
}

// Farthest point sampling: one block per batch
__global__ void fps_kernel(int* p_idx, const float* pts, int N, int S) {
  __shared__ float dist[3072];
  __shared__ float rv[256];
  __shared__ int   ri[256];
  __shared__ int   s_last;
  const int b = blockIdx.x;
  const int tid = threadIdx.x;
  const float* P = pts + (size_t)b * N * 3;

  for (int n = tid; n < N; n += blockDim.x) dist[n] = 1e10f;
  if (tid == 0) { p_idx[(size_t)b * S] = 0; s_last = 0; }
  __syncthreads();

  for (int it = 1; it < S; ++it) {
    const int last = s_last;
    const float lx = P[last * 3 + 0];
    const float ly = P[last * 3 + 1];
    const float lz = P[last * 3 + 2];
    float bv = -1.0f;
    int bi = 0;
    for (int n = tid; n < N; n += blockDim.x) {
      float dx = P[n * 3 + 0] - lx;
      float dy = P[n * 3 + 1] - ly;
      float dz = P[n * 3 + 2] - lz;
      float d = dx * dx + dy * dy + dz * dz;
      float dn = fminf(dist[n], d);
      dist[n] = dn;
      if (dn > bv) { bv = dn; bi = n; }
    }
    rv[tid] = bv; ri[tid] = bi;
    __syncthreads();
    for (int sred = 128; sred > 0; sred >>= 1) {
      if (tid < sred) {
        float ov = rv[tid + sred]; int oi = ri[tid + sred];
        if (ov > rv[tid] || (ov == rv[tid] && oi < ri[tid])) { rv[tid] = ov; ri[tid] = oi; }
      }
      __syncthreads();
    }
    if (tid == 0) { p_idx[(size_t)b * S + it] = ri[0]; s_last = ri[0]; }
    __syncthreads();
  }
}

__global__ void gather_pts_kernel(float* out, const float* pts, const int* p_idx,
                                  int N, int S, int total) {
  int i = blockIdx.x * blockDim.x + threadIdx.x;
  if (i >= total) return;
  int s = i % S;
  int b = i / S;
  int src = p_idx[(size_t)b * S + s];
  for (int c = 0; c < 3; ++c)
    out[((size_t)b * S + s) * 3 + c] = pts[((size_t)b * N + src) * 3 + c];
}

// f(B,2C,S): rows [0,C)=center gather, rows [C,2C)=max over 16 neighbors
__global__ void eps_feat_kernel(float* f, const float* feat, const int* p_idx,
                                const int* pn_idx, int C, int N, int S, int total) {
  int i = blockIdx.x * blockDim.x + threadIdx.x;
  if (i >= total) return;
  int s = i % S;
  int t = i / S;
  int c = t % C;
  int b = t / C;
  const float* fb = feat + ((size_t)b * C + c) * N;
  int pc = p_idx[(size_t)b * S + s];
  const int* nb = pn_idx + ((size_t)b * S + s) * 16;
  float mx = -3.4e38f;
  for (int j = 0; j < 16; ++j) mx = fmaxf(mx, fb[nb[j]]);
  f[((size_t)b * (2 * C) + c) * S + s] = fb[pc];
  f[((size_t)b * (2 * C) + C + c) * S + s] = mx;
}

// 3-NN weights for interpolation
__global__ void knn3_kernel(int* idx3, float* w3, const float* src, const float* tgt,
                            int S, int Nt, int total) {
  int i = blockIdx.x * blockDim.x + threadIdx.x;
  if (i >= total) return;
  int n = i % Nt;
  int b = i / Nt;
  const float* T = tgt + ((size_t)b * Nt + n) * 3;
  const float qx = T[0], qy = T[1], qz = T[2];
  const float* P = src + (size_t)b * S * 3;
  float d0 = 3.4e38f, d1 = 3.4e38f, d2 = 3.4e38f;
  int i0 = 0, i1 = 0, i2 = 0;
  for (int m = 0; m < S; ++m) {
    float dx = P[m * 3 + 0] - qx;
    float dy = P[m * 3 + 1] - qy;
    float dz = P[m * 3 + 2] - qz;
    float d = dx * dx + dy * dy + dz * dz;
    if (d < d2) {
      if (d < d0)      { d2 = d1; i2 = i1; d1 = d0; i1 = i0; d0 = d; i0 = m; }
      else if (d < d1) { d2 = d1; i2 = i1; d1 = d; i1 = m; }
      else             { d2 = d; i2 = m; }
    }
  }
  float w0 = 1.0f / fmaxf(sqrtf(fmaxf(d0, 0.0f)), 1e-10f);
  float w1 = 1.0f / fmaxf(sqrtf(fmaxf(d1, 0.0f)), 1e-10f);
  float w2 = 1.0f / fmaxf(sqrtf(fmaxf(d2, 0.0f)), 1e-10f);
  float inv = 1.0f / (w0 + w1 + w2);
  size_t base = ((size_t)b * Nt + n) * 3;
  idx3[base + 0] = i0; idx3[base + 1] = i1; idx3[base + 2] = i2;
  w3[base + 0] = w0 * inv; w3[base + 1] = w1 * inv; w3[base + 2] = w2 * inv;
}

// dst(B,Ctot,Nt) rows [row0,row0+Csrc) = interp of feat(B,Csrc,Ns)
__global__ void interp_apply_kernel(float* dst, const float* feat, const int* idx3,
                                    const float* w3, int Csrc, int Ns, int Nt,
                                    int Ctot, int row0, int total) {
  int i = blockIdx.x * blockDim.x + threadIdx.x;
  if (i >= total) return;
  int n = i % Nt;
  int t = i / Nt;
  int c = t % Csrc;
  int b = t / Csrc;
  const float* fb = feat + ((size_t)b * Csrc + c) * Ns;
  size_t base = ((size_t)b * Nt + n) * 3;
  float s = w3[base + 0] * fb[idx3[base + 0]]
          + w3[base + 1] * fb[idx3[base + 1]]
          + w3[base + 2] * fb[idx3[base + 2]];
  dst[((size_t)b * Ctot + row0 + c) * Nt + n] = s;
}

// ---------------------------------------------------------------------------
// Host-side structures & helpers
// ---------------------------------------------------------------------------
struct SAParams {
  const float *w1, *b1, *w2, *b2, *w3, *b3, *wA, *wB, *bB, *wo, *bo;
};
struct SKParams {
  SAParams sams[2];
  const float *fc_w, *fc_b, *fcs_w[2], *fcs_b[2];
};
struct UnitParams {
  const float* conv1_w;
  SKParams blocks[2];
  const float *conv2_w, *conv_res_w;
};
struct Scratch {
  float *xcur, *xa, *res, *feas0, *feas1;
  float *x1s, *x2s, *x3s, *wcat, *wa, *wb, *wsum;
  float *fea_s, *fea_z, *vecs;
};

static void launch_gemm(hipStream_t st, const float* W, const float* X,
                        const float* bias, const float* res, float* Y,
                        int O, int C, int N, int act) {
  dim3 grid(cdiv(N, 64), cdiv(O, 16), BB);
  dim3 blk(128);
  switch (N) {
    case 3072: gemm_wmma_kernel<3072><<<grid, blk, 0, st>>>(W, X, bias, res, Y, O, C, N, act); break;
    case 1536: gemm_wmma_kernel<1536><<<grid, blk, 0, st>>>(W, X, bias, res, Y, O, C, N, act); break;
    case 768:  gemm_wmma_kernel<768><<<grid, blk, 0, st>>>(W, X, bias, res, Y, O, C, N, act); break;
    case 384:  gemm_wmma_kernel<384><<<grid, blk, 0, st>>>(W, X, bias, res, Y, O, C, N, act); break;
    default:   gemm_wmma_kernel<0><<<grid, blk, 0, st>>>(W, X, bias, res, Y, O, C, N, act); break;
  }
}

static void launch_gemm_gather(hipStream_t st, const float* W, const float* XA,
                               const int* idx, const float* bias, float* Y,
                               int O, int C, int N, int K, int act) {
  dim3 grid(cdiv(K * N, 64), cdiv(O, 16), BB);
  dim3 blk(128);
  switch (N) {
    case 3072: gemm_gather_kernel<3072><<<grid, blk, 0, st>>>(W, XA, idx, bias, Y, O, C, N, K, act); break;
    case 1536: gemm_gather_kernel<1536><<<grid, blk, 0, st>>>(W, XA, idx, bias, Y, O, C, N, K, act); break;
    case 768:  gemm_gather_kernel<768><<<grid, blk, 0, st>>>(W, XA, idx, bias, Y, O, C, N, K, act); break;
    case 384:  gemm_gather_kernel<384><<<grid, blk, 0, st>>>(W, XA, idx, bias, Y, O, C, N, K, act); break;
    default:   gemm_gather_kernel<0><<<grid, blk, 0, st>>>(W, XA, idx, bias, Y, O, C, N, K, act); break;
  }
}

#define EL_LAUNCH(kern, n, ...) \
  kern<<<cdiv((n), 256), 256, 0, st>>>(__VA_ARGS__)

static void run_sk_sa(hipStream_t st, const SKParams& p, int C, int N,
                      const int* idx10, const int* idx20, Scratch& S) {
  const int ks[2] = {10, 20};
  const int rel = C / 16, mid = C / 4, ms = mid / SHARE;
  const int d = C / 2;  // == max(C/2,32) for C>=64

  EL_LAUNCH(relu_kernel, BB * C * N, S.xa, S.xcur, BB * C * N);

  float* feas[2] = {S.feas0, S.feas1};
  for (int i = 0; i < 2; ++i) {
    const SAParams& a = p.sams[i];
    const int k = ks[i];
    const int* idx = (i == 0) ? idx10 : idx20;
    launch_gemm(st, a.w1, S.xa, a.b1, nullptr, S.x1s, rel, C, N, ACT_NONE);
    launch_gemm_gather(st, a.w2, S.xa, idx, a.b2, S.x2s, rel, C, N, k, ACT_NONE);
    launch_gemm_gather(st, a.w3, S.xa, idx, a.b3, S.x3s, mid, C, N, k, ACT_NONE);
    const int P = rel * (k + 1);
    EL_LAUNCH(pack_wcat_kernel, BB * P * N, S.wcat, S.x1s, S.x2s, rel, k, N, BB * P * N);
    launch_gemm(st, a.wA, S.wcat, nullptr, nullptr, S.wa, ms, P, N, ACT_RELU);
    launch_gemm(st, a.wB, S.wa, a.bB, nullptr, S.wb, k * ms, ms, N, ACT_NONE);
    EL_LAUNCH(weighted_sum_kernel, BB * mid * N, S.wsum, S.wb, S.x3s, mid, ms, k, N,
              BB * mid * N);
    launch_gemm(st, a.wo, S.wsum, a.bo, S.xcur, feas[i], C, mid, N, ACT_RELU);
  }

  EL_LAUNCH(mean_over_n_kernel, BB * C, S.fea_s, S.feas0, S.feas1, N, BB * C);
  EL_LAUNCH(linear_kernel, BB * d, S.fea_z, S.fea_s, p.fc_w, p.fc_b, C, d, d, 0,
            ACT_NONE, BB * d);
  EL_LAUNCH(linear_kernel, BB * C, S.vecs, S.fea_z, p.fcs_w[0], p.fcs_b[0], d, C,
            2 * C, 0, ACT_NONE, BB * C);
  EL_LAUNCH(linear_kernel, BB * C, S.vecs, S.fea_z, p.fcs_w[1], p.fcs_b[1], d, C,
            2 * C, C, ACT_NONE, BB * C);
  EL_LAUNCH(sk_combine_kernel, BB * C * N, S.xcur, S.feas0, S.feas1, S.vecs, C, N,
            BB * C * N);
}

static void run_unit(hipStream_t st, const UnitParams& u, const float* feat_in,
                     int cin, int C, int N, const int* idx10, const int* idx20,
                     float* out, Scratch& S) {
  launch_gemm(st, u.conv1_w, feat_in, nullptr, nullptr, S.xcur, C, cin, N, ACT_NONE);
  for (int bl = 0; bl < 2; ++bl)
    run_sk_sa(st, u.blocks[bl], C, N, idx10, idx20, S);
  EL_LAUNCH(relu_kernel, BB * C * N, S.xa, S.xcur, BB * C * N);
  launch_gemm(st, u.conv_res_w, feat_in, nullptr, nullptr, S.res, C, cin, N, ACT_NONE);
  // out = relu( conv2 * relu(x) + conv_res * feat_in )  (caller relu fused here)
  launch_gemm(st, u.conv2_w, S.xa, nullptr, S.res, out, C, C, N, ACT_RELU);
}

// ---------------------------------------------------------------------------
// Entry point
// ---------------------------------------------------------------------------
extern "C" void kernel_launch(void* const* d_in, const int* in_sizes, int n_in,
                              void* d_out, int out_size, void* d_ws, size_t ws_size,
                              hipStream_t stream) {
  (void)in_sizes; (void)n_in; (void)out_size; (void)ws_size;
  hipStream_t st = stream;

  const int Ns[4] = {3072, 1536, 768, 384};
  const int Cs[4] = {64, 128, 256, 512};
  const int ks[2] = {10, 20};

  // -- parse flattened inputs in setup_inputs() insertion order ----------------
  int cur = 0;
  auto nxt = [&]() -> const float* { return (const float*)d_in[cur++]; };
  const float* features = nxt();
  UnitParams U[4];
  for (int u = 0; u < 4; ++u) {
    U[u].conv1_w = nxt();
    for (int bl = 0; bl < 2; ++bl) {
      for (int s = 0; s < 2; ++s) {
        SAParams& a = U[u].blocks[bl].sams[s];
        a.w1 = nxt(); a.b1 = nxt();
        a.w2 = nxt(); a.b2 = nxt();
        a.w3 = nxt(); a.b3 = nxt();
        a.wA = nxt(); a.wB = nxt(); a.bB = nxt();
        a.wo = nxt(); a.bo = nxt();
      }
      SKParams& sk = U[u].blocks[bl];
      sk.fc_w = nxt(); sk.fc_b = nxt();
      sk.fcs_w[0] = nxt(); sk.fcs_w[1] = nxt();
      sk.fcs_b[0] = nxt(); sk.fcs_b[1] = nxt();
    }
    U[u].conv2_w = nxt();
    U[u].conv_res_w = nxt();
  }
  const float* conv5_w = nxt(); const float* conv5_b = nxt();
  const float* fc1_w = nxt();  const float* fc1_b = nxt();
  const float* fc2_w = nxt();  const float* fc2_b = nxt();
  const float* conv6_w = nxt(); const float* conv6_b = nxt();
  const float* conv7_w = nxt(); const float* conv7_b = nxt();
  const float* conv8_w = nxt(); const float* conv8_b = nxt();
  const float* conv9_w = nxt(); const float* conv9_b = nxt();
  const float* convo_w = nxt(); const float* convo_b = nxt();

  // -- workspace layout --------------------------------------------------------
  size_t off = 0;
  auto alloc = [&](size_t elems) -> void* {
    void* p = (char*)d_ws + off;
    off += ((elems * 4 + 255) / 256) * 256;
    return p;
  };
  const size_t FB = (size_t)BB * 64 * 3072;  // C*N identical across levels

  float* pt[4];
  for (int l = 0; l < 4; ++l) pt[l] = (float*)alloc((size_t)BB * Ns[l] * 3);
  int* idxK[4][2];
  for (int l = 0; l < 4; ++l)
    for (int i = 0; i < 2; ++i)
      idxK[l][i] = (int*)alloc((size_t)BB * Ns[l] * ks[i]);
  float* xl[4];
  for (int l = 0; l < 4; ++l) xl[l] = (float*)alloc(FB);
  float* unit_in = (float*)alloc(FB);
  int* p_idx = (int*)alloc((size_t)BB * 1536);
  int* pn_idx = (int*)alloc((size_t)BB * 1536 * 16);

  const size_t arena = off;

  // per-unit scratch (arena view A)
  size_t acur = arena;
  auto aalloc = [&](size_t elems) -> void* {
    void* p = (char*)d_ws + acur;
    acur += ((elems * 4 + 255) / 256) * 256;
    return p;
  };
  Scratch S;
  S.xcur = (float*)aalloc(FB);
  S.xa   = (float*)aalloc(FB);
  S.res  = (float*)aalloc(FB);
  S.feas0 = (float*)aalloc(FB);
  S.feas1 = (float*)aalloc(FB);
  S.x1s  = (float*)aalloc((size_t)BB * 12288);        // rel*N per batch
  S.x2s  = (float*)aalloc((size_t)BB * 12288 * 20);   // rel*K*N
  S.x3s  = (float*)aalloc((size_t)BB * 49152 * 20);   // mid*K*N
  S.wcat = (float*)aalloc((size_t)BB * 12288 * 21);   // rel*(K+1)*N
  S.wa   = (float*)aalloc((size_t)BB * 6144);         // ms*N
  S.wb   = (float*)aalloc((size_t)BB * 6144 * 20);    // K*ms*N
  S.wsum = (float*)aalloc((size_t)BB * 49152);        // mid*N
  S.fea_s = (float*)aalloc((size_t)BB * 512);
  S.fea_z = (float*)aalloc((size_t)BB * 256);
  S.vecs  = (float*)aalloc((size_t)BB * 2 * 512);

  // decoder scratch (arena view B, reuses unit-scratch region after units done)
  size_t dcur = arena;
  auto dalloc = [&](size_t elems) -> void* {
    void* p = (char*)d_ws + dcur;
    dcur += ((elems * 4 + 255) / 256) * 256;
    return p;
  };
  float* g5   = (float*)dalloc((size_t)BB * 1024 * 384);
  float* gmax = (float*)dalloc((size_t)BB * 1024);
  float* fz1  = (float*)dalloc((size_t)BB * 512);
  float* gvec = (float*)dalloc((size_t)BB * 1024);
  float* cat6 = (float*)dalloc((size_t)BB * 1536 * 384);
  float* c6o  = (float*)dalloc((size_t)BB * 512 * 384);
  float* cat7 = (float*)dalloc((size_t)BB * 768 * 768);
  float* c7o  = (float*)dalloc((size_t)BB * 256 * 768);
  float* cat8 = (float*)dalloc((size_t)BB * 384 * 1536);
  float* c8o  = (float*)dalloc((size_t)BB * 128 * 1536);
  float* cat9 = (float*)dalloc((size_t)BB * 192 * 3072);
  float* c9o  = (float*)dalloc((size_t)BB * 64 * 3072);
  int*   i3idx = (int*)dalloc((size_t)BB * 3072 * 3);
  float* i3w   = (float*)dalloc((size_t)BB * 3072 * 3);

  // -- encoder -----------------------------------------------------------------
  EL_LAUNCH(transpose_pts_kernel, BB * Ns[0], pt[0], features, Ns[0], BB * Ns[0]);

  const float* feat_in = features;
  int cin = 3;
  for (int l = 0; l < 4; ++l) {
    const int C = Cs[l], N = Ns[l];
    // self-kNN on this level's points
    for (int i = 0; i < 2; ++i)
      EL_LAUNCH(knn_query_kernel, BB * N, idxK[l][i], pt[l], pt[l], N, N, ks[i],
                BB * N);
    run_unit(st, U[l], feat_in, cin, C, N, idxK[l][0], idxK[l][1], xl[l], S);

    if (l < 3) {
      const int Sn = Ns[l + 1];
      fps_kernel<<<BB, 256, 0, st>>>(p_idx, pt[l], N, Sn);
      EL_LAUNCH(gather_pts_kernel, BB * Sn, pt[l + 1], pt[l], p_idx, N, Sn, BB * Sn);
      EL_LAUNCH(knn_query_kernel, BB * Sn, pn_idx, pt[l], pt[l + 1], N, Sn, 16,
                BB * Sn);
      EL_LAUNCH(eps_feat_kernel, BB * C * Sn, unit_in, xl[l], p_idx, pn_idx, C, N,
                Sn, BB * C * Sn);
      feat_in = unit_in;
      cin = 2 * C;
    }
  }

  // -- decoder -----------------------------------------------------------------
  launch_gemm(st, conv5_w, xl[3], conv5_b, nullptr, g5, 1024, 512, 384, ACT_NONE);
  EL_LAUNCH(max_over_n_kernel, BB * 1024, gmax, g5, 384, BB * 1024);
  EL_LAUNCH(linear_kernel, BB * 512, fz1, gmax, fc1_w, fc1_b, 1024, 512, 512, 0,
            ACT_RELU, BB * 512);
  EL_LAUNCH(linear_kernel, BB * 1024, gvec, fz1, fc2_w, fc2_b, 512, 1024, 1024, 0,
            ACT_RELU, BB * 1024);

  EL_LAUNCH(broadcast_rows_kernel, BB * 1024 * 384, cat6, gvec, 1536, 0, 1024, 384,
            BB * 1024 * 384);
  EL_LAUNCH(copy_rows_kernel, BB * 512 * 384, cat6, xl[3], 1536, 1024, 512, 384,
            BB * 512 * 384);
  launch_gemm(st, conv6_w, cat6, conv6_b, nullptr, c6o, 512, 1536, 384, ACT_RELU);

  // up: pt4 -> pt3
  EL_LAUNCH(knn3_kernel, BB * 768, i3idx, i3w, pt[3], pt[2], 384, 768, BB * 768);
  EL_LAUNCH(interp_apply_kernel, BB * 512 * 768, cat7, c6o, i3idx, i3w, 512, 384,
            768, 768, 0, BB * 512 * 768);
  EL_LAUNCH(copy_rows_kernel, BB * 256 * 768, cat7, xl[2], 768, 512, 256, 768,
            BB * 256 * 768);
  launch_gemm(st, conv7_w, cat7, conv7_b, nullptr, c7o, 256, 768, 768, ACT_RELU);

  // up: pt3 -> pt2
  EL_LAUNCH(knn3_kernel, BB * 1536, i3idx, i3w, pt[2], pt[1], 768, 1536, BB * 1536);
  EL_LAUNCH(interp_apply_kernel, BB * 256 * 1536, cat8, c7o, i3idx, i3w, 256, 768,
            1536, 384, 0, BB * 256 * 1536);
  EL_LAUNCH(copy_rows_kernel, BB * 128 * 1536, cat8, xl[1], 384, 256, 128, 1536,
            BB * 128 * 1536);
  launch_gemm(st, conv8_w, cat8, conv8_b, nullptr, c8o, 128, 384, 1536, ACT_RELU);

  // up: pt2 -> pt1
  EL_LAUNCH(knn3_kernel, BB * 3072, i3idx, i3w, pt[1], pt[0], 1536, 3072, BB * 3072);
  EL_LAUNCH(interp_apply_kernel, BB * 128 * 3072, cat9, c8o, i3idx, i3w, 128, 1536,
            3072, 192, 0, BB * 128 * 3072);
  EL_LAUNCH(copy_rows_kernel, BB * 64 * 3072, cat9, xl[0], 192, 128, 64, 3072,
            BB * 64 * 3072);
  launch_gemm(st, conv9_w, cat9, conv9_b, nullptr, c9o, 64, 192, 3072, ACT_RELU);

  launch_gemm(st, convo_w, c9o, convo_b, nullptr, (float*)d_out, 64, 64, 3072,
              ACT_NONE);
}